// MP2Vec_15075335209513
// MI455X (gfx1250) — compile-verified
//
#include <hip/hip_runtime.h>
#include <hip/hip_bf16.h>

typedef __attribute__((ext_vector_type(2))) float v2f;
typedef __attribute__((ext_vector_type(4))) float v4f;
typedef __attribute__((ext_vector_type(8))) float v8f;
typedef __attribute__((ext_vector_type(4))) int   v4i;

typedef __attribute__((address_space(1))) v4i* g_v4i;   // global int4*
typedef __attribute__((address_space(3))) v4i* l_v4i;   // LDS int4*

#define EMBED_DIM 128
#define W_POS 5
#define W_NEG 10
#define NUM_TYPES 4
#define ROWS_PER_BLOCK 4          // 128 threads = 4 wave32, one batch-row per wave
#define ROW_DW 132                // 128 dwords + 4 pad -> bank stride 4 (conflict-free)
#define WAVE_LDS_DW (16 * ROW_DW + EMBED_DIM)   // 16 sample rows + 1 start row

#if __has_builtin(__builtin_amdgcn_global_load_async_to_lds_b128)
#define HAVE_ASYNC_LDS 1
#else
#define HAVE_ASYNC_LDS 0
#endif

__device__ __forceinline__ void wait_async_zero() {
#if __has_builtin(__builtin_amdgcn_s_wait_asynccnt)
    __builtin_amdgcn_s_wait_asynccnt(0);
    __asm__ volatile("" ::: "memory");
#else
    __asm__ volatile("s_wait_asynccnt 0x0" ::: "memory");
#endif
}

__global__ __launch_bounds__(128) void mp2v_wmma_kernel(
    const float* __restrict__ start_embeds,   // [NODE_SIZE, 128]
    const float* __restrict__ end_embeds,     // [NODE_SIZE, 4, 128]
    const int*   __restrict__ start_node,     // [B]
    const int*   __restrict__ pos_samples,    // [B, 5]
    const int*   __restrict__ neg_samples,    // [B, 10]
    const int*   __restrict__ node_types,     // [NODE_SIZE]
    float*       __restrict__ out,            // [B]
    int Btot)
{
    __shared__ float lds[ROWS_PER_BLOCK * WAVE_LDS_DW];

    const int lane = threadIdx.x & 31;
    const int row  = blockIdx.x * ROWS_PER_BLOCK + (threadIdx.x >> 5);
    if (row >= Btot) return;   // wave-uniform guard (no divergence before WMMA)

    float* __restrict__ wlds = lds + (threadIdx.x >> 5) * WAVE_LDS_DW;  // wave-private

    // ---- gather indices ----
    const int sn = start_node[row];
    const int ty = node_types[sn];
    const float* __restrict__ srow = start_embeds + (size_t)sn * EMBED_DIM;

    // slot owned by this lane pair: 0..4 pos, 5..14 neg, 15 dummy
    const int m = lane & 15;
    int sample;
    if (m < W_POS)              sample = pos_samples[row * W_POS + m];
    else if (m < W_POS + W_NEG) sample = neg_samples[row * W_NEG + (m - W_POS)];
    else                        sample = 0;   // dummy slot, result discarded

    // ---- stage start row + 16 sample rows into LDS, fully coalesced ----
    // __shfl(sample, r) lowers to v_readlane -> SGPR-uniform row base, so each
    // row transfer is base + lane*16: every cacheline touched exactly once.
#if HAVE_ASYNC_LDS
    // CDNA5 async DMA: global -> LDS directly, tracked by ASYNCcnt. No VGPR
    // round-trip, no loadcnt->dscnt serialization.
    __builtin_amdgcn_global_load_async_to_lds_b128(
        (g_v4i)(srow + lane * 4),
        (l_v4i)(wlds + 16 * ROW_DW + lane * 4), 0, 0);
#pragma unroll
    for (int r = 0; r < 16; ++r) {
        const int sr = __shfl(sample, r, 32);
        const float* __restrict__ erow =
            end_embeds + ((size_t)sr * NUM_TYPES + (size_t)ty) * EMBED_DIM;
        __builtin_amdgcn_global_load_async_to_lds_b128(
            (g_v4i)(erow + lane * 4),
            (l_v4i)(wlds + r * ROW_DW + lane * 4), 0, 0);
    }
    wait_async_zero();   // all 17 row DMAs visible in LDS
#else
    {
        v4f s4 = *(const v4f*)(srow + lane * 4);
        *(v4f*)(wlds + 16 * ROW_DW + lane * 4) = s4;
    }
#pragma unroll
    for (int r = 0; r < 16; ++r) {
        const int sr = __shfl(sample, r, 32);
        const float* __restrict__ erow =
            end_embeds + ((size_t)sr * NUM_TYPES + (size_t)ty) * EMBED_DIM;
        v4f d = *(const v4f*)(erow + lane * 4);
        *(v4f*)(wlds + r * ROW_DW + lane * 4) = d;
    }
#endif
    // Region is wave-private: same-wave DS ordering suffices, no barrier.

    // ---- D = A(16 samples x K) x B(start vec replicated over N), K=128 ----
    // f32 WMMA 16x16x4 layouts: lanes 0-15 own K={k0,k0+1}, lanes 16-31 K={k0+2,k0+3}
    const int koffBase = (lane >> 4) * 2;
    const float* __restrict__ arow = wlds + (lane & 15) * ROW_DW;  // bank-stride 4: conflict-free
    const float* __restrict__ sld  = wlds + 16 * ROW_DW;           // broadcast reads

    v8f c = {};
#pragma unroll 8
    for (int k0 = 0; k0 < EMBED_DIM; k0 += 4) {
        const int ko = k0 + koffBase;
        v2f a  = *(const v2f*)(arow + ko);   // ds_load_b64, all 64 banks hit once
        v2f bb = *(const v2f*)(sld + ko);    // ds_load_b64 broadcast (2 unique addrs)
        c = __builtin_amdgcn_wmma_f32_16x16x4_f32(
                /*neg_a=*/false, a, /*neg_b=*/false, bb,
                /*c_mod=*/(short)0, c, /*reuse_a=*/false, /*reuse_b=*/false);
    }

    // ---- epilogue: C VGPR j holds dot for sample (j + 8*(lane>=16)), any column ----
    const int mbase = (lane >> 4) * 8;
    float pos_sum = 0.0f, neg_sum = 0.0f;
#pragma unroll
    for (int j = 0; j < 8; ++j) {
        const float d  = c[j];
        const int   mm = mbase + j;
        if (mm < W_POS) {
            pos_sum += __logf(1.0f / (1.0f + __expf(-d)) + 1e-15f);        // log(sigmoid(d)+eps)
        } else if (mm < W_POS + W_NEG) {
            neg_sum += __logf(1.0f / (1.0f + __expf(d)) + 1e-15f);         // log(1-sigmoid(d)+eps)
        }
        // mm == 15: dummy, skip
    }
    pos_sum += __shfl_xor(pos_sum, 16, 32);
    neg_sum += __shfl_xor(neg_sum, 16, 32);

    if (lane == 0) {
        out[row] = -pos_sum * (1.0f / W_POS) - neg_sum * (1.0f / W_NEG);
    }
}

extern "C" void kernel_launch(void* const* d_in, const int* in_sizes, int n_in,
                              void* d_out, int out_size, void* d_ws, size_t ws_size,
                              hipStream_t stream) {
    const float* start_embeds = (const float*)d_in[0];
    const float* end_embeds   = (const float*)d_in[1];
    const int*   start_node   = (const int*)  d_in[2];
    const int*   pos_samples  = (const int*)  d_in[3];
    const int*   neg_samples  = (const int*)  d_in[4];
    const int*   node_types   = (const int*)  d_in[5];
    float* out = (float*)d_out;

    const int Btot = in_sizes[2];  // start_node has B elements
    const int grid = (Btot + ROWS_PER_BLOCK - 1) / ROWS_PER_BLOCK;

    mp2v_wmma_kernel<<<grid, 128, 0, stream>>>(
        start_embeds, end_embeds, start_node, pos_samples, neg_samples,
        node_types, out, Btot);
}